// GlobalAllPooling_59107339927781
// MI455X (gfx1250) — compile-verified
//
#include <hip/hip_runtime.h>
#include <hip/hip_bf16.h>
#include <stdint.h>

// ---------------------------------------------------------------------------
// Segment pooling (mean | max | add) for sorted batch indices.
// Memory-bound: ~205 MB streamed once -> ~8.8us floor at 23.3 TB/s.
// CDNA5 path: async global->LDS copies (ASYNCcnt) as a depth-4 pipeline.
// ---------------------------------------------------------------------------

#define PIPE_DEPTH 4

#if defined(__HIP_DEVICE_COMPILE__) && defined(__gfx1250__) && \
    __has_builtin(__builtin_amdgcn_global_load_async_to_lds_b128) && \
    __has_builtin(__builtin_amdgcn_s_wait_asynccnt)
#define USE_ASYNC_LDS 1
#else
#define USE_ASYNC_LDS 0
#endif

// 128-bit payload type the async-LDS builtins expect (v4i32).
typedef int v4i __attribute__((vector_size(4 * sizeof(int))));
typedef __attribute__((address_space(1))) v4i v4i_as1;   // global
typedef __attribute__((address_space(3))) v4i v4i_as3;   // LDS

// ---------------------------------------------------------------------------
// Kernel 1: segment boundaries via binary search (batch is sorted).
// bounds[g] = first i with batch[i] >= g ; bounds[G] = N.
// ---------------------------------------------------------------------------
__global__ void seg_bounds_kernel(const int* __restrict__ batch, int N, int G,
                                  int* __restrict__ bounds) {
  int g = blockIdx.x * blockDim.x + threadIdx.x;
  if (g > G) return;
  if (g == G) { bounds[G] = N; return; }
  int lo = 0, hi = N;
  while (lo < hi) {
    int mid = (lo + hi) >> 1;
    if (batch[mid] < g) lo = mid + 1; else hi = mid;
  }
  bounds[g] = lo;
}

// ---------------------------------------------------------------------------
// Kernel 2: per-(segment, row-chunk) partial sum + max over rows.
// blockDim.x == C/4 threads; thread t owns channels [4t, 4t+4).
// Rows stream through a depth-PIPE_DEPTH async LDS pipeline; each lane's
// async copy targets its own LDS slot, so only s_wait_asynccnt is needed
// (no workgroup barrier).
// ---------------------------------------------------------------------------
__global__ void __launch_bounds__(256) seg_reduce_kernel(
    const float* __restrict__ x, const int* __restrict__ bounds,
    int C, int splits,
    float* __restrict__ psum, float* __restrict__ pmax) {
  extern __shared__ float lds[];  // PIPE_DEPTH * C floats

  const int g   = blockIdx.x;
  const int k   = blockIdx.y;
  const int tid = threadIdx.x;
  const int c   = tid * 4;

  const int s   = bounds[g];
  const int e   = bounds[g + 1];
  const int len = e - s;
  const int r0  = s + (int)(((long long)len * k) / splits);
  const int r1  = s + (int)(((long long)len * (k + 1)) / splits);

  float s0 = 0.f, s1 = 0.f, s2 = 0.f, s3 = 0.f;
  float m0 = -__builtin_inff(), m1 = -__builtin_inff();
  float m2 = -__builtin_inff(), m3 = -__builtin_inff();

#if USE_ASYNC_LDS
  {
    int r = r0;
    // Prologue: fill the pipeline.
    int nissue = r1 - r0; if (nissue > PIPE_DEPTH) nissue = PIPE_DEPTH;
    for (int i = 0; i < nissue; ++i) {
      const float* src = x + (size_t)(r0 + i) * C + c;
      float* dst = &lds[(size_t)i * C + c];
      __builtin_amdgcn_global_load_async_to_lds_b128(
          (v4i_as1*)src, (v4i_as3*)dst, 0, 0);
    }
    int slot = 0;
    // Steady state: consume one row, issue one row, <= PIPE_DEPTH in flight.
    while (r + PIPE_DEPTH < r1) {
      __builtin_amdgcn_s_wait_asynccnt(PIPE_DEPTH - 1);  // row r's copy done
      float4 v = *(const float4*)&lds[(size_t)slot * C + c];
      s0 += v.x; s1 += v.y; s2 += v.z; s3 += v.w;
      m0 = fmaxf(m0, v.x); m1 = fmaxf(m1, v.y);
      m2 = fmaxf(m2, v.z); m3 = fmaxf(m3, v.w);
      const float* src = x + (size_t)(r + PIPE_DEPTH) * C + c;
      float* dst = &lds[(size_t)slot * C + c];
      __builtin_amdgcn_global_load_async_to_lds_b128(
          (v4i_as1*)src, (v4i_as3*)dst, 0, 0);
      slot = (slot + 1 == PIPE_DEPTH) ? 0 : slot + 1;
      ++r;
    }
    // Epilogue: drain.
    __builtin_amdgcn_s_wait_asynccnt(0);
    while (r < r1) {
      float4 v = *(const float4*)&lds[(size_t)slot * C + c];
      s0 += v.x; s1 += v.y; s2 += v.z; s3 += v.w;
      m0 = fmaxf(m0, v.x); m1 = fmaxf(m1, v.y);
      m2 = fmaxf(m2, v.z); m3 = fmaxf(m3, v.w);
      slot = (slot + 1 == PIPE_DEPTH) ? 0 : slot + 1;
      ++r;
    }
  }
#else
  for (int r = r0; r < r1; ++r) {
    float4 v = *(const float4*)(x + (size_t)r * C + c);
    s0 += v.x; s1 += v.y; s2 += v.z; s3 += v.w;
    m0 = fmaxf(m0, v.x); m1 = fmaxf(m1, v.y);
    m2 = fmaxf(m2, v.z); m3 = fmaxf(m3, v.w);
  }
#endif

  const size_t base = ((size_t)g * splits + k) * C + c;
  float4 sv; sv.x = s0; sv.y = s1; sv.z = s2; sv.w = s3;
  float4 mv; mv.x = m0; mv.y = m1; mv.z = m2; mv.w = m3;
  *(float4*)&psum[base] = sv;
  *(float4*)&pmax[base] = mv;
}

// ---------------------------------------------------------------------------
// Kernel 3: fold partials, compute mean / max / add and write [G, 3C].
// One thread per (g, 4-channel group).
// ---------------------------------------------------------------------------
__global__ void seg_combine_kernel(const float* __restrict__ psum,
                                   const float* __restrict__ pmax,
                                   const int* __restrict__ bounds,
                                   int C, int G, int splits,
                                   float* __restrict__ out) {
  const int cq  = C >> 2;
  const int idx = blockIdx.x * blockDim.x + threadIdx.x;
  if (idx >= G * cq) return;
  const int g = idx / cq;
  const int c = (idx - g * cq) * 4;

  float s0 = 0.f, s1 = 0.f, s2 = 0.f, s3 = 0.f;
  float m0 = -__builtin_inff(), m1 = -__builtin_inff();
  float m2 = -__builtin_inff(), m3 = -__builtin_inff();
  for (int k = 0; k < splits; ++k) {
    const size_t base = ((size_t)g * splits + k) * C + c;
    float4 sv = *(const float4*)&psum[base];
    float4 mv = *(const float4*)&pmax[base];
    s0 += sv.x; s1 += sv.y; s2 += sv.z; s3 += sv.w;
    m0 = fmaxf(m0, mv.x); m1 = fmaxf(m1, mv.y);
    m2 = fmaxf(m2, mv.z); m3 = fmaxf(m3, mv.w);
  }
  const int cnt = bounds[g + 1] - bounds[g];
  const float inv = 1.0f / (float)(cnt > 1 ? cnt : 1);
  if (cnt <= 0) { m0 = 0.f; m1 = 0.f; m2 = 0.f; m3 = 0.f; }

  const size_t orow = (size_t)g * 3 * C;
  float4 mean; mean.x = s0 * inv; mean.y = s1 * inv; mean.z = s2 * inv; mean.w = s3 * inv;
  float4 mx;   mx.x = m0; mx.y = m1; mx.z = m2; mx.w = m3;
  float4 sm;   sm.x = s0; sm.y = s1; sm.z = s2; sm.w = s3;
  *(float4*)&out[orow + c]           = mean;
  *(float4*)&out[orow + C + c]       = mx;
  *(float4*)&out[orow + 2 * C + c]   = sm;
}

// ---------------------------------------------------------------------------
// Host launcher. Inputs: d_in[0]=x (fp32, N*C), d_in[1]=batch (int, N),
// d_in[2]=num_graphs (scalar). Output: [G, 3C] fp32.
// ---------------------------------------------------------------------------
extern "C" void kernel_launch(void* const* d_in, const int* in_sizes, int n_in,
                              void* d_out, int out_size, void* d_ws, size_t ws_size,
                              hipStream_t stream) {
  const float* x     = (const float*)d_in[0];
  const int*   batch = (const int*)d_in[1];
  (void)n_in;

  const int N = in_sizes[1];
  const int C = in_sizes[0] / N;           // 512
  const int G = out_size / (3 * C);        // 128

  // Workspace layout: [bounds: (G+1) ints, 256B-aligned pad] [psum] [pmax]
  int* bounds = (int*)d_ws;
  const size_t off = (((size_t)(G + 1) * sizeof(int)) + 255) & ~(size_t)255;

  int splits = 16;
  while (splits > 1 &&
         off + 2ull * splits * G * C * sizeof(float) > ws_size) {
    splits >>= 1;
  }
  float* psum = (float*)((char*)d_ws + off);
  float* pmax = psum + (size_t)splits * G * C;

  seg_bounds_kernel<<<(G + 1 + 255) / 256, 256, 0, stream>>>(batch, N, G, bounds);

  dim3 grid(G, splits);
  const int threads = C / 4;                       // 128 for C=512
  const size_t shmem = (size_t)PIPE_DEPTH * C * sizeof(float);  // 8 KB
  seg_reduce_kernel<<<grid, threads, shmem, stream>>>(x, bounds, C, splits,
                                                      psum, pmax);

  const int total = G * (C / 4);
  seg_combine_kernel<<<(total + 127) / 128, 128, 0, stream>>>(
      psum, pmax, bounds, C, G, splits, (float*)d_out);
}